// LSEAggregator_1073741824126
// MI455X (gfx1250) — compile-verified
//
#include <hip/hip_runtime.h>
#include <math.h>

// Segment logsumexp: N rows x 256 cols f32, G sorted segments -> [G, 256] f32.
// Memory-bound (512MB read @ 23.3TB/s ~ 22us). One block per segment,
// thread t owns column t. Data streamed via CDNA5 async global->LDS copies
// (GLOBAL_LOAD_ASYNC_TO_LDS_B128, ASYNCcnt) with double buffering.

#define HCOLS            256                     // H, also blockDim.x
#define TILE_ROWS        8                       // rows per LDS tile
#define ROW_BYTES        (HCOLS * 4)             // 1024 B
#define TILE_BYTES       (TILE_ROWS * ROW_BYTES) // 8192 B
#define CHUNKS_PER_THREAD (TILE_BYTES / (HCOLS * 16)) // 16B chunks each thread copies
static_assert(CHUNKS_PER_THREAD == 2, "wait count hard-coded below");

// One 16-byte async copy: global (per-lane 64-bit vaddr, GV mode) -> LDS byte addr.
__device__ __forceinline__ void async_copy_b128(unsigned lds_byte_addr, const void* gaddr) {
    asm volatile("global_load_async_to_lds_b128 %0, %1, off"
                 :
                 : "v"(lds_byte_addr), "v"(gaddr)
                 : "memory");
}

__device__ __forceinline__ void wait_asynccnt_le2() {
    asm volatile("s_wait_asynccnt 0x2" ::: "memory");
}
__device__ __forceinline__ void wait_asynccnt_le0() {
    asm volatile("s_wait_asynccnt 0x0" ::: "memory");
}

extern "C" __global__ __launch_bounds__(HCOLS)
void lse_segment_kernel(const float* __restrict__ x,
                        const int*   __restrict__ ids,
                        float*       __restrict__ out,
                        int N, int G) {
    __shared__ float smem[2][TILE_ROWS * HCOLS];   // 2 x 8KB double buffer
    __shared__ int   sBounds[2];

    const int g   = blockIdx.x;
    const int tid = threadIdx.x;

    // pair_ids is sorted: binary search segment bounds once, broadcast via LDS.
    if (tid == 0) {
        int lo = 0, hi = N;
        while (lo < hi) { int mid = (lo + hi) >> 1; if (ids[mid] <  g) lo = mid + 1; else hi = mid; }
        const int s0 = lo;
        hi = N;
        while (lo < hi) { int mid = (lo + hi) >> 1; if (ids[mid] <= g) lo = mid + 1; else hi = mid; }
        sBounds[0] = s0;
        sBounds[1] = lo;
    }
    __syncthreads();
    const int seg_start = sBounds[0];
    const int seg_end   = sBounds[1];
    const int rows      = seg_end - seg_start;

    if (rows <= 0) {  // uniform across block; reference guarantees non-empty
        out[(size_t)g * HCOLS + tid] = -INFINITY;
        return;
    }

    const int   ntiles = (rows + TILE_ROWS - 1) / TILE_ROWS;
    const char* base   = (const char*)x;
    // LDS byte offsets of the two buffers (generic->LDS offset = low 32 bits).
    const unsigned lds_base0 = (unsigned)(unsigned long long)(&smem[0][0]);
    const unsigned lds_base1 = (unsigned)(unsigned long long)(&smem[1][0]);

    // Issue all async copies for one tile. Each of 256 threads copies
    // CHUNKS_PER_THREAD 16B chunks => 2 wave-wide async instructions per wave.
    auto issue_tile = [&](int tile, int buf) {
        const unsigned lbase = buf ? lds_base1 : lds_base0;
#pragma unroll
        for (int c = 0; c < CHUNKS_PER_THREAD; ++c) {
            const int lin   = c * HCOLS + tid;      // 16B-chunk index within tile
            const int lrow  = lin >> 6;             // 64 chunks per 1KB row
            const int inrow = (lin & 63) << 4;      // byte offset within row
            int grow = tile * TILE_ROWS + lrow;
            if (grow > rows - 1) grow = rows - 1;   // clamp partial tile (unread garbage)
            const void* ga = base + ((size_t)(seg_start + grow) * ROW_BYTES + (size_t)inrow);
            async_copy_b128(lbase + (unsigned)(lin << 4), ga);
        }
    };

    // Online (one-pass) logsumexp state for this thread's column.
    float m = -INFINITY;
    float s = 0.0f;

    issue_tile(0, 0);
    for (int t = 0; t < ntiles; ++t) {
        const int  buf     = t & 1;
        const bool hasNext = (t + 1) < ntiles;
        if (hasNext) issue_tile(t + 1, buf ^ 1);   // prefetch next tile into other buffer

        // Async loads complete in order: waiting to <=2 retires this wave's
        // copies for tile t while tile t+1's 2 copies stay in flight.
        if (hasNext) wait_asynccnt_le2();
        else         wait_asynccnt_le0();
        __syncthreads();   // all waves' portions of tile t now visible in LDS

        const int rem  = rows - t * TILE_ROWS;
        const int rlim = rem < TILE_ROWS ? rem : TILE_ROWS;
        for (int r = 0; r < rlim; ++r) {
            const float v = smem[buf][r * HCOLS + tid];
            if (v <= m) {
                s += __expf(v - m);
            } else {                 // rare: new running max, rescale sum
                s = s * __expf(m - v) + 1.0f;
                m = v;
            }
        }
        __syncthreads();   // done reading buf before iteration t+1 overwrites it
    }

    out[(size_t)g * HCOLS + tid] = m + __logf(s);
}

extern "C" void kernel_launch(void* const* d_in, const int* in_sizes, int n_in,
                              void* d_out, int out_size, void* d_ws, size_t ws_size,
                              hipStream_t stream) {
    const float* seq_rep  = (const float*)d_in[0];
    const int*   pair_ids = (const int*)d_in[1];
    float*       out      = (float*)d_out;

    const int N = in_sizes[1];            // number of rows (== len(pair_ids))
    const int G = out_size / HCOLS;       // number of segments

    hipLaunchKernelGGL(lse_segment_kernel, dim3(G), dim3(HCOLS), 0, stream,
                       seq_rep, pair_ids, out, N, G);
}